// GCN_65094524338333
// MI455X (gfx1250) — compile-verified
//
#include <hip/hip_runtime.h>

#define IN_F  128
#define HID_F 64
#define OUT_F 32

typedef float v2f __attribute__((ext_vector_type(2)));
typedef float v8f __attribute__((ext_vector_type(8)));

#if defined(__HIP_DEVICE_COMPILE__) && __has_builtin(__builtin_amdgcn_tensor_load_to_lds)
#define HAVE_TDM 1
#else
#define HAVE_TDM 0
#endif

#if HAVE_TDM
typedef unsigned int u32x4 __attribute__((ext_vector_type(4)));
typedef int          i32x4 __attribute__((ext_vector_type(4)));
typedef int          i32x8 __attribute__((ext_vector_type(8)));

// Issue a TDM DMA of a dense row-major [dim1 x dim0] f32 tensor into LDS.
// D# layout per CDNA5 ISA ch.8 (group0 128b, group1 256b; groups 2/3 zero for 2D).
__device__ __forceinline__ void tdm_load_2d_to_lds(unsigned lds_off, const void* gaddr,
                                                   unsigned dim0, unsigned dim1) {
  unsigned long long ga = (unsigned long long)gaddr;
  u32x4 g0;
  g0[0] = 1u;                                               // count=1, is_restore=0
  g0[1] = lds_off;                                          // lds_addr
  g0[2] = (unsigned)(ga & 0xffffffffu);                     // global_addr[31:0]
  g0[3] = (unsigned)((ga >> 32) & 0x01ffffffu) | (2u << 30);// global_addr[56:32], type=2
  i32x8 g1;
  g1[0] = (int)(2u << 16);                                  // data_size=2 (4B), mask=0
  g1[1] = (int)((dim0 & 0xffffu) << 16);                    // tensor_dim0[15:0]
  g1[2] = (int)(((dim0 >> 16) & 0xffffu) | ((dim1 & 0xffffu) << 16)); // dim0 hi | dim1 lo
  g1[3] = (int)(((dim1 >> 16) & 0xffffu) | ((dim0 & 0xffffu) << 16)); // dim1 hi | tile_dim0
  g1[4] = (int)(dim1 & 0xffffu);                            // tile_dim1 | tile_dim2=0
  g1[5] = (int)dim0;                                        // tensor_dim0_stride[31:0]
  g1[6] = 0;
  g1[7] = 0;
  i32x4 z4 = {0, 0, 0, 0};
#if __clang_major__ >= 23
  i32x8 z8 = {0, 0, 0, 0, 0, 0, 0, 0};
  __builtin_amdgcn_tensor_load_to_lds(g0, g1, z4, z4, z8, 0);
#else
  __builtin_amdgcn_tensor_load_to_lds(g0, g1, z4, z4, 0);
#endif
  __builtin_amdgcn_s_wait_tensorcnt(0);
}
#endif

// ---------------- graph preprocessing ----------------

__global__ void k_edges_to_i32(const long long* __restrict__ ei,
                               int* __restrict__ s32, int* __restrict__ d32, int E) {
  int t = blockIdx.x * blockDim.x + threadIdx.x;
  if (t < E) {
    s32[t] = (int)ei[t];
    d32[t] = (int)ei[(size_t)E + t];
  }
}

__global__ void k_init_deg(float* __restrict__ deg, int n) {
  int t = blockIdx.x * blockDim.x + threadIdx.x;
  if (t < n) deg[t] = 1.0f;               // self-loop contributes 1
}

__global__ void k_count_deg(const int* __restrict__ d32, float* __restrict__ deg, int E) {
  int t = blockIdx.x * blockDim.x + threadIdx.x;
  if (t < E) atomicAdd(&deg[d32[t]], 1.0f);
}

__global__ void k_rsqrt_inplace(float* __restrict__ deg, int n) {
  int t = blockIdx.x * blockDim.x + threadIdx.x;
  if (t < n) deg[t] = rsqrtf(deg[t]);     // deg >= 1 guaranteed
}

__global__ void k_edge_norm(const int* __restrict__ s32, const int* __restrict__ d32,
                            const float* __restrict__ dis, float* __restrict__ nrm, int E) {
  int t = blockIdx.x * blockDim.x + threadIdx.x;
  if (t < E) nrm[t] = dis[s32[t]] * dis[d32[t]];
}

// ---------------- dense GEMM via f32 WMMA with TDM-staged B in LDS ----------------
// C[M x N] = A[M x K] * B[K x N], row-major, N = NT*16, K = KK.
// Block: 8 waves; wave w computes rows [ (blk*8+w)*16 , +16 ) x full N (NT tiles).
// B is staged once per block into LDS (TDM DMA when available).
template <int NT, int KK>
__global__ __launch_bounds__(256) void k_gemm_wmma_lds(const float* __restrict__ A,
                                                       const float* __restrict__ B,
                                                       float* __restrict__ C, int M) {
  constexpr int N = NT * 16;
  __shared__ float sB[KK * N];

#if HAVE_TDM
  if ((threadIdx.x >> 5) == 0) {          // one wave issues the DMA + waits
    unsigned lds_off = (unsigned)(unsigned long long)(void*)sB;
    tdm_load_2d_to_lds(lds_off, B, (unsigned)N, (unsigned)KK);
  }
#else
  for (int i = threadIdx.x; i < KK * N; i += 256) sB[i] = B[i];
#endif
  __syncthreads();

  const int lane  = threadIdx.x & 31;
  const int wave  = threadIdx.x >> 5;
  const int mtile = blockIdx.x * 8 + wave;
  if (mtile * 16 >= M) return;            // wave-uniform exit (after barrier)

  const int l16 = lane & 15;
  const int hi  = lane >> 4;              // 0: lanes 0-15, 1: lanes 16-31
  const int kh  = hi << 1;                // 0 or 2
  const int mrow = mtile * 16 + l16;

  const float* Ap = A + (size_t)mrow * KK + kh;

  v8f acc[NT];
#pragma unroll
  for (int nt = 0; nt < NT; ++nt) acc[nt] = (v8f){};

  for (int k = 0; k < KK; k += 4) {
    v2f a;
    a.x = Ap[k];
    a.y = Ap[k + 1];
    const float* bs = sB + (k + kh) * N + l16;
#pragma unroll
    for (int nt = 0; nt < NT; ++nt) {
      v2f b;
      b.x = bs[nt * 16];                  // row k+kh
      b.y = bs[N + nt * 16];              // row k+kh+1
      acc[nt] = __builtin_amdgcn_wmma_f32_16x16x4_f32(
          false, a, false, b, (short)0, acc[nt], false, false);
    }
  }

  const int rbase = mtile * 16 + (hi << 3);
#pragma unroll
  for (int nt = 0; nt < NT; ++nt) {
    float* Cp = C + (size_t)rbase * N + nt * 16 + l16;
#pragma unroll
    for (int v = 0; v < 8; ++v) Cp[(size_t)v * N] = acc[nt][v];
  }
}

// ---------------- elementwise helpers ----------------

__global__ void k_zero(float* __restrict__ p, long long n4) {  // n4 = count of float4
  long long t = (long long)blockIdx.x * blockDim.x + threadIdx.x;
  if (t < n4) ((float4*)p)[t] = make_float4(0.f, 0.f, 0.f, 0.f);
}

__global__ void k_bias_relu(float* __restrict__ h, const float* __restrict__ b, int n) {
  int t = blockIdx.x * blockDim.x + threadIdx.x;
  if (t < n) {
    float v = h[t] + b[t & (HID_F - 1)];
    h[t] = v > 0.f ? v : 0.f;
  }
}

__global__ void k_out_init(float* __restrict__ o, const float* __restrict__ b, int n) {
  int t = blockIdx.x * blockDim.x + threadIdx.x;
  if (t < n) o[t] = b[t & (OUT_F - 1)];
}

// ---------------- sparse aggregation (gather * norm, scatter-add) ----------------
// One thread = one float4 of one (edge | self-loop). fqShift = log2(F/4).
// All element offsets fit in 32 bits (N*F <= 3.2M), so index math stays i32.
__global__ void k_aggregate(const float* __restrict__ H, const int* __restrict__ s32,
                            const int* __restrict__ d32, const float* __restrict__ nrm,
                            const float* __restrict__ dis, float* __restrict__ out,
                            int E, int n, int F, int fqShift) {
  long long t = (long long)blockIdx.x * blockDim.x + threadIdx.x;
  long long total = ((long long)(E + n)) << fqShift;
  if (t >= total) return;
  int e  = (int)(t >> fqShift);
  int f0 = ((int)t & ((1 << fqShift) - 1)) << 2;
  int s, d;
  float w;
  if (e < E) {
    s = s32[e]; d = d32[e]; w = nrm[e];
  } else {                                  // self-loop, norm = dis[i]^2
    s = e - E; d = s;
    float di = dis[s];
    w = di * di;
  }
  const float4 hv = *(const float4*)(H + (s * F + f0));
  float* o = out + (d * F + f0);
  atomicAdd(o + 0, hv.x * w);
  atomicAdd(o + 1, hv.y * w);
  atomicAdd(o + 2, hv.z * w);
  atomicAdd(o + 3, hv.w * w);
}

// ---------------- launcher ----------------

static inline int gs(long long total, int tb) { return (int)((total + tb - 1) / tb); }

extern "C" void kernel_launch(void* const* d_in, const int* in_sizes, int n_in,
                              void* d_out, int out_size, void* d_ws, size_t ws_size,
                              hipStream_t stream) {
  (void)n_in; (void)out_size; (void)ws_size;
  const float*     x  = (const float*)d_in[0];
  const long long* ei = (const long long*)d_in[1];
  const float*     W1 = (const float*)d_in[2];
  const float*     b1 = (const float*)d_in[3];
  const float*     W2 = (const float*)d_in[4];
  const float*     b2 = (const float*)d_in[5];

  const int n = in_sizes[0] / IN_F;   // 50000 (multiple of 16)
  const int E = in_sizes[1] / 2;      // 800000

  // workspace carve-up (floats)
  float* ws  = (float*)d_ws;
  float* dis = ws;                               // n        (deg -> rsqrt in place)
  int*   s32 = (int*)(dis + n);                  // E
  int*   d32 = s32 + E;                          // E
  float* nrm = (float*)(d32 + E);                // E
  float* H1  = nrm + E;                          // n*HID
  float* AG1 = H1 + (size_t)n * HID_F;           // n*HID (agg1 -> relu h in place)
  float* H2  = AG1 + (size_t)n * HID_F;          // n*OUT
  float* out = (float*)d_out;

  const int TB = 256;
  const int mtiles = n / 16;                     // 3125
  const int gemm_blocks = (mtiles + 7) / 8;      // 391

  // normalization coefficients
  k_edges_to_i32<<<gs(E, TB), TB, 0, stream>>>(ei, s32, d32, E);
  k_init_deg<<<gs(n, TB), TB, 0, stream>>>(dis, n);
  k_count_deg<<<gs(E, TB), TB, 0, stream>>>(d32, dis, E);
  k_rsqrt_inplace<<<gs(n, TB), TB, 0, stream>>>(dis, n);
  k_edge_norm<<<gs(E, TB), TB, 0, stream>>>(s32, d32, dis, nrm, E);

  // layer 1: H1 = x @ W1 ; AG1 = A_hat @ H1 ; h = relu(AG1 + b1)
  k_gemm_wmma_lds<HID_F / 16, IN_F><<<gemm_blocks, 256, 0, stream>>>(x, W1, H1, n);
  k_zero<<<gs((long long)n * HID_F / 4, TB), TB, 0, stream>>>(AG1, (long long)n * HID_F / 4);
  k_aggregate<<<gs(((long long)(E + n)) * (HID_F / 4), TB), TB, 0, stream>>>(
      H1, s32, d32, nrm, dis, AG1, E, n, HID_F, 4);
  k_bias_relu<<<gs((long long)n * HID_F, TB), TB, 0, stream>>>(AG1, b1, n * HID_F);

  // layer 2: H2 = h @ W2 ; out = A_hat @ H2 + b2
  k_gemm_wmma_lds<OUT_F / 16, HID_F><<<gemm_blocks, 256, 0, stream>>>(AG1, W2, H2, n);
  k_out_init<<<gs((long long)n * OUT_F, TB), TB, 0, stream>>>(out, b2, n * OUT_F);
  k_aggregate<<<gs(((long long)(E + n)) * (OUT_F / 4), TB), TB, 0, stream>>>(
      H2, s32, d32, nrm, dis, out, E, n, OUT_F, 3);
}